// CANConv_8289286881714
// MI455X (gfx1250) — compile-verified
//
#include <hip/hip_runtime.h>
#include <math.h>

typedef __attribute__((ext_vector_type(2))) float v2f;
typedef __attribute__((ext_vector_type(8))) float v8f;

#define B  4
#define CIN 32
#define COUT 32
#define H  64
#define W  64
#define S  4096        // H*W
#define KC 32
#define AREA 9
#define F  288         // CIN*AREA
#define ITERS 10

// ---------------- workspace layout (byte offsets) ----------------
#define OFF_FEAT   0u                       // B*S*CIN floats  = 2,097,152 B
#define OFF_IDX    2097152u                 // B*S ints        =    65,536 B
#define OFF_PCENT  2162688u                 // B*KC*F floats   =   147,456 B
#define OFF_WCIN   2310144u                 // B*KC*32 floats
#define OFF_WAREA  2326528u                 // B*KC*9  floats
#define OFF_WCOUT  2331136u                 // B*KC*32 floats
#define OFF_BIAS   2347520u                 // B*KC*32 floats
#define OFF_BASET  2363904u                 // 32*288 floats (transposed base)
#define OFF_MP     2400768u                 // B*S*F floats   = 18,874,368 B

// ---------------------------------------------------------------
// 1) feat[b,p,c] = mean over 3x3 zero-padded neighborhood of x[b,c]
__global__ __launch_bounds__(256) void feat_kernel(const float* __restrict__ x,
                                                   float* __restrict__ feat) {
    int t = blockIdx.x * 256 + threadIdx.x;      // 0 .. B*CIN*S-1
    int b = t >> 17;
    int c = (t >> 12) & 31;
    int p = t & 4095;
    int y = p >> 6, xx = p & 63;
    const float* xb = x + (((size_t)(b << 5) + c) << 12);
    float s = 0.f;
    #pragma unroll
    for (int di = -1; di <= 1; ++di)
        #pragma unroll
        for (int dj = -1; dj <= 1; ++dj) {
            int yy = y + di, xj = xx + dj;
            if ((unsigned)yy < 64u && (unsigned)xj < 64u) s += xb[(yy << 6) + xj];
        }
    feat[(((size_t)((b << 12) | p)) << 5) | c] = s / 9.0f;
}

// ---------------------------------------------------------------
// 2) KMeans: one workgroup per batch. Centroids + segment sums in LDS.
__global__ __launch_bounds__(256) void kmeans_kernel(const float* __restrict__ feat,
                                                     int* __restrict__ idx_ws,
                                                     int* __restrict__ idx_out) {
    int b = blockIdx.x;
    const float* Fm = feat + (size_t)b * S * CIN;
    __shared__ float cent[KC][CIN];
    __shared__ float sums[KC][CIN];
    __shared__ float cnt[KC];
    __shared__ float c2[KC];
    int tid = threadIdx.x;

    for (int i = tid; i < KC * CIN; i += 256) {
        int k = i >> 5, c = i & 31;
        cent[k][c] = Fm[(size_t)(k * 128) * CIN + c];   // strided init feat[k*(S/KC)]
    }
    __syncthreads();

    for (int it = 0; it < ITERS; ++it) {
        if (tid < KC) {
            float s = 0.f;
            #pragma unroll
            for (int c = 0; c < CIN; ++c) s += cent[tid][c] * cent[tid][c];
            c2[tid] = s;
            cnt[tid] = 0.f;
        }
        for (int i = tid; i < KC * CIN; i += 256) sums[i >> 5][i & 31] = 0.f;
        __syncthreads();

        for (int p = tid; p < S; p += 256) {
            float fv[CIN];
            const float* fp = Fm + (size_t)p * CIN;
            #pragma unroll
            for (int c = 0; c < CIN; ++c) fv[c] = fp[c];
            int best = 0; float bd = 3.4e38f;
            for (int k = 0; k < KC; ++k) {
                float dot = 0.f;
                #pragma unroll
                for (int c = 0; c < CIN; ++c) dot += fv[c] * cent[k][c];
                float d = c2[k] - 2.0f * dot;        // f2 term constant per-pixel
                if (d < bd) { bd = d; best = k; }    // first-min tie-break
            }
            atomicAdd(&cnt[best], 1.0f);
            #pragma unroll
            for (int c = 0; c < CIN; ++c) atomicAdd(&sums[best][c], fv[c]);
        }
        __syncthreads();
        for (int i = tid; i < KC * CIN; i += 256) {
            int k = i >> 5, c = i & 31;
            float n = cnt[k];
            if (n > 0.f) cent[k][c] = sums[k][c] / n;  // max(cnt,1)==n when n>0
        }
        __syncthreads();
    }
    // final assignment with converged centroids
    if (tid < KC) {
        float s = 0.f;
        #pragma unroll
        for (int c = 0; c < CIN; ++c) s += cent[tid][c] * cent[tid][c];
        c2[tid] = s;
    }
    __syncthreads();
    for (int p = tid; p < S; p += 256) {
        const float* fp = Fm + (size_t)p * CIN;
        float fv[CIN];
        #pragma unroll
        for (int c = 0; c < CIN; ++c) fv[c] = fp[c];
        int best = 0; float bd = 3.4e38f;
        for (int k = 0; k < KC; ++k) {
            float dot = 0.f;
            #pragma unroll
            for (int c = 0; c < CIN; ++c) dot += fv[c] * cent[k][c];
            float d = c2[k] - 2.0f * dot;
            if (d < bd) { bd = d; best = k; }
        }
        idx_ws[b * S + p] = best;
        idx_out[b * S + p] = best;
    }
}

// ---------------------------------------------------------------
// 3) per-cluster mean of 288-dim patches (patches recomputed on the fly)
__global__ __launch_bounds__(256) void pcent_kernel(const float* __restrict__ x,
                                                    const int* __restrict__ idx,
                                                    float* __restrict__ pcent) {
    int b = blockIdx.x;
    __shared__ float sums[KC * F];    // 36,864 B
    __shared__ float cnt[KC];
    int tid = threadIdx.x;
    for (int i = tid; i < KC * F; i += 256) sums[i] = 0.f;
    if (tid < KC) cnt[tid] = 0.f;
    __syncthreads();

    for (int p = tid; p < S; p += 256) {
        int k = idx[b * S + p];
        atomicAdd(&cnt[k], 1.0f);
        int y = p >> 6, xx = p & 63;
        for (int c = 0; c < CIN; ++c) {
            const float* xb = x + (((size_t)(b << 5) + c) << 12);
            #pragma unroll
            for (int a = 0; a < AREA; ++a) {
                int di = a / 3 - 1, dj = a % 3 - 1;
                int yy = y + di, xj = xx + dj;
                float v = 0.f;
                if ((unsigned)yy < 64u && (unsigned)xj < 64u) v = xb[(yy << 6) + xj];
                atomicAdd(&sums[k * F + c * AREA + a], v);
            }
        }
    }
    __syncthreads();
    for (int i = tid; i < KC * F; i += 256) {
        int k = i / F;
        pcent[(size_t)b * KC * F + i] = sums[i] / fmaxf(cnt[k], 1.0f);
    }
}

// ---------------------------------------------------------------
// 4) tiny MLPs per (b,k) centroid row: one wave per row
__global__ __launch_bounds__(32) void mlp_kernel(const float* __restrict__ pcent,
    const float* Wh1, const float* bh1, const float* Wh2, const float* bh2,
    const float* Wa,  const float* ba,  const float* Wc,  const float* bc,
    const float* Wo,  const float* bo,
    const float* Wb1, const float* bb1, const float* Wb2, const float* bb2,
    const float* Wb3, const float* bb3,
    float* __restrict__ w_cin, float* __restrict__ w_area,
    float* __restrict__ w_cout, float* __restrict__ biasv) {
    int row = blockIdx.x;                 // b*KC + k
    const float* cen = pcent + (size_t)row * F;
    __shared__ float h1[16], h2[16], g1[16], g2[16];
    int t = threadIdx.x;
    if (t < 16) {
        float s = bh1[t], sb = bb1[t];
        for (int f = 0; f < F; ++f) {
            float cv = cen[f];
            s  += cv * Wh1[f * 16 + t];
            sb += cv * Wb1[f * 16 + t];
        }
        h1[t] = fmaxf(s, 0.f);
        g1[t] = fmaxf(sb, 0.f);
    }
    __syncthreads();
    if (t < 16) {
        float s = bh2[t], sb = bb2[t];
        #pragma unroll
        for (int j = 0; j < 16; ++j) { s += h1[j] * Wh2[j * 16 + t]; sb += g1[j] * Wb2[j * 16 + t]; }
        h2[t] = fmaxf(s, 0.f);
        g2[t] = fmaxf(sb, 0.f);
    }
    __syncthreads();
    {   // w_cin head (32) - sigmoid
        float s = bc[t];
        #pragma unroll
        for (int j = 0; j < 16; ++j) s += h2[j] * Wc[j * 32 + t];
        w_cin[row * 32 + t] = 1.0f / (1.0f + expf(-s));
    }
    {   // w_cout head (32) - sigmoid
        float s = bo[t];
        #pragma unroll
        for (int j = 0; j < 16; ++j) s += h2[j] * Wo[j * 32 + t];
        w_cout[row * 32 + t] = 1.0f / (1.0f + expf(-s));
    }
    if (t < AREA) {   // w_area head (9) - sigmoid
        float s = ba[t];
        #pragma unroll
        for (int j = 0; j < 16; ++j) s += h2[j] * Wa[j * AREA + t];
        w_area[row * AREA + t] = 1.0f / (1.0f + expf(-s));
    }
    {   // bias head (32) - linear
        float s = bb3[t];
        #pragma unroll
        for (int j = 0; j < 16; ++j) s += g2[j] * Wb3[j * 32 + t];
        biasv[row * 32 + t] = s;
    }
}

// ---------------------------------------------------------------
// 5) transpose shared base kernel [288][32] -> baseT [32][288]
__global__ __launch_bounds__(256) void transpose_kernel(const float* __restrict__ kern,
                                                        float* __restrict__ baseT) {
    int t = blockIdx.x * 256 + threadIdx.x;   // 0..9215
    int o = t / F, f = t % F;
    baseT[o * F + f] = kern[f * COUT + o];
}

// ---------------------------------------------------------------
// 6) modulated patches: mp[p][f] = patch(p,f) * w_cin[idx[p],c] * w_area[idx[p],a]
__global__ __launch_bounds__(288) void prep_kernel(const float* __restrict__ x,
                                                   const int* __restrict__ idx,
                                                   const float* __restrict__ w_cin,
                                                   const float* __restrict__ w_area,
                                                   float* __restrict__ mp) {
    int p = blockIdx.x;            // 0..16383
    int f = threadIdx.x;           // 0..287
    int b = p >> 12, yx = p & 4095;
    int y = yx >> 6, xx = yx & 63;
    int k = idx[p];
    int c = f / AREA, a = f % AREA;
    int di = a / 3 - 1, dj = a % 3 - 1;
    int yy = y + di, xj = xx + dj;
    float v = 0.f;
    if ((unsigned)yy < 64u && (unsigned)xj < 64u)
        v = x[(((size_t)(b << 5) + c) << 12) + (yy << 6) + xj];
    int bk = (b << 5) + k;
    v *= w_cin[(bk << 5) + c] * w_area[bk * AREA + a];
    mp[(size_t)p * F + f] = v;
}

// ---------------------------------------------------------------
// 7) WMMA GEMM: [16384,288] @ baseT^T[288,32], post-scale by w_cout + bias
//    one wave = 16-pixel M-tile, both 16-wide N halves; f32 16x16x4 WMMA
__global__ __launch_bounds__(128) void gemm_kernel(const float* __restrict__ mp,
                                                   const float* __restrict__ baseT,
                                                   const int* __restrict__ idx,
                                                   const float* __restrict__ w_cout,
                                                   const float* __restrict__ biasv,
                                                   float* __restrict__ out) {
    int wave = threadIdx.x >> 5;
    int lane = threadIdx.x & 31;
    int tile = blockIdx.x * 4 + wave;       // 0..1023
    int p0 = tile << 4;                     // first pixel of M-tile
    int m  = lane & 15;                     // A row / C col index
    int hi = lane >> 4;                     // lane-half selects K pair / M offset

    const float* arow = mp + (size_t)(p0 + m) * F;
    const float* brow0 = baseT + (size_t)m * F;          // column n=m of base
    const float* brow1 = baseT + (size_t)(m + 16) * F;   // column n=m+16

    v8f acc0 = {};
    v8f acc1 = {};
    for (int kb = 0; kb < F; kb += 4) {
        int k0 = kb + (hi << 1);
        v2f a  = *(const v2f*)(arow  + k0);
        v2f b0 = *(const v2f*)(brow0 + k0);
        v2f b1 = *(const v2f*)(brow1 + k0);
        acc0 = __builtin_amdgcn_wmma_f32_16x16x4_f32(false, a, false, b0,
                                                     (short)0, acc0, false, false);
        acc1 = __builtin_amdgcn_wmma_f32_16x16x4_f32(false, a, false, b1,
                                                     (short)0, acc1, false, false);
    }

    // C/D layout: VGPR r holds M = r + 8*hi, N = lane&15
    #pragma unroll
    for (int r = 0; r < 8; ++r) {
        int q = p0 + r + (hi << 3);          // pixel
        int b = q >> 12, yx = q & 4095;
        int k = idx[q];
        int base_ko = ((b << 5) + k) << 5;
        int o0 = m;
        float v0 = acc0[r] * w_cout[base_ko + o0]      + biasv[base_ko + o0];
        float v1 = acc1[r] * w_cout[base_ko + o0 + 16] + biasv[base_ko + o0 + 16];
        out[(((size_t)(b << 5) + o0)      << 12) + yx] = v0;
        out[(((size_t)(b << 5) + o0 + 16) << 12) + yx] = v1;
    }
}

// ---------------------------------------------------------------
extern "C" void kernel_launch(void* const* d_in, const int* in_sizes, int n_in,
                              void* d_out, int out_size, void* d_ws, size_t ws_size,
                              hipStream_t stream) {
    const float* x   = (const float*)d_in[0];
    const float* Wh1 = (const float*)d_in[1];  const float* bh1 = (const float*)d_in[2];
    const float* Wh2 = (const float*)d_in[3];  const float* bh2 = (const float*)d_in[4];
    const float* Wa  = (const float*)d_in[5];  const float* ba  = (const float*)d_in[6];
    const float* Wc  = (const float*)d_in[7];  const float* bc  = (const float*)d_in[8];
    const float* Wo  = (const float*)d_in[9];  const float* bo  = (const float*)d_in[10];
    const float* kernels = (const float*)d_in[11];
    const float* Wb1 = (const float*)d_in[12]; const float* bb1 = (const float*)d_in[13];
    const float* Wb2 = (const float*)d_in[14]; const float* bb2 = (const float*)d_in[15];
    const float* Wb3 = (const float*)d_in[16]; const float* bb3 = (const float*)d_in[17];

    char* ws = (char*)d_ws;
    float* feat   = (float*)(ws + OFF_FEAT);
    int*   idxw   = (int*)  (ws + OFF_IDX);
    float* pcent  = (float*)(ws + OFF_PCENT);
    float* w_cin  = (float*)(ws + OFF_WCIN);
    float* w_area = (float*)(ws + OFF_WAREA);
    float* w_cout = (float*)(ws + OFF_WCOUT);
    float* biasv  = (float*)(ws + OFF_BIAS);
    float* baseT  = (float*)(ws + OFF_BASET);
    float* mp     = (float*)(ws + OFF_MP);

    float* out     = (float*)d_out;                       // [4,32,64,64]
    int*   idx_out = (int*)d_out + B * COUT * S;          // [4,4096] int32 tail

    feat_kernel<<<B * CIN * S / 256, 256, 0, stream>>>(x, feat);
    kmeans_kernel<<<B, 256, 0, stream>>>(feat, idxw, idx_out);
    pcent_kernel<<<B, 256, 0, stream>>>(x, idxw, pcent);
    mlp_kernel<<<B * KC, 32, 0, stream>>>(pcent,
        Wh1, bh1, Wh2, bh2, Wa, ba, Wc, bc, Wo, bo,
        Wb1, bb1, Wb2, bb2, Wb3, bb3,
        w_cin, w_area, w_cout, biasv);
    transpose_kernel<<<(COUT * F + 255) / 256, 256, 0, stream>>>(kernels, baseT);
    prep_kernel<<<B * S, F, 0, stream>>>(x, idxw, w_cin, w_area, mp);
    gemm_kernel<<<(B * S / 16) / 4, 128, 0, stream>>>(mp, baseT, idxw, w_cout, biasv, out);
}